// GCN_74964359185003
// MI455X (gfx1250) — compile-verified
//
#include <hip/hip_runtime.h>
#include <hip/hip_bf16.h>
#include <math.h>

typedef __attribute__((ext_vector_type(2))) float v2f;
typedef __attribute__((ext_vector_type(8))) float v8f;

// ---------------------------------------------------------------------------
// Degree / normalization
// ---------------------------------------------------------------------------
__global__ void k_init_deg(float* __restrict__ deg, int n) {
    int i = blockIdx.x * blockDim.x + threadIdx.x;
    if (i < n) deg[i] = 1.0f;  // self-loop contributes 1
}

__global__ void k_deg_edges(const long long* __restrict__ dst,
                            float* __restrict__ deg, int E) {
    int stride = gridDim.x * blockDim.x;
    for (int e = blockIdx.x * blockDim.x + threadIdx.x; e < E; e += stride) {
        __builtin_prefetch(&dst[e + stride], 0, 1);   // global_prefetch_b8
        atomicAdd(&deg[(int)dst[e]], 1.0f);
    }
}

__global__ void k_dinv(const float* __restrict__ deg,
                       float* __restrict__ dinv, int n) {
    int i = blockIdx.x * blockDim.x + threadIdx.x;
    if (i < n) dinv[i] = rsqrtf(deg[i]);
}

// ---------------------------------------------------------------------------
// Layer 1 dense transform: h1[N,16] = x[N,2] @ W1[2,16]  (K padded 2 -> 4)
// One wave per 16-row tile, single V_WMMA_F32_16X16X4_F32.
// A layout (16x4 f32): lanes 0-15 -> K=0(v0),K=1(v1); lanes 16-31 -> K=2,K=3.
// C/D layout: VGPR r, lanes 0-15 -> M=r, lanes 16-31 -> M=r+8; N = lane&15.
// Branchless operand construction: clamped-address loads + selects so EXEC
// stays all-ones into the WMMA and no saveexec regions are generated.
// ---------------------------------------------------------------------------
__global__ void k_gemm1_wmma(const float* __restrict__ x,
                             const float* __restrict__ W1,
                             float* __restrict__ h1, int n) {
    int gtid = blockIdx.x * blockDim.x + threadIdx.x;
    int wave = gtid >> 5;
    int lane = threadIdx.x & 31;
    int numTiles = (n + 15) >> 4;
    if (wave >= numTiles) return;       // wave-uniform exit

    int r0 = wave << 4;
    int m  = lane & 15;
    int hi = lane >> 4;                 // 0: K=0,1   1: K=2,3 (zero pad)
    int row = r0 + m;
    bool full = (r0 + 16 <= n);         // wave-uniform
    int rr = row < n ? row : (n - 1);   // clamp, load always valid

    float2 xv = *(const float2*)(x + (size_t)rr * 2);
    float w0 = W1[m];                   // W1[0][m]
    float w1 = W1[16 + m];              // W1[1][m]

    bool alive = (hi == 0) && (row < n);
    v2f a, b;
    a.x = alive ? xv.x : 0.0f;
    a.y = alive ? xv.y : 0.0f;
    b.x = (hi == 0) ? w0 : 0.0f;
    b.y = (hi == 0) ? w1 : 0.0f;

    v8f c = {};
    c = __builtin_amdgcn_wmma_f32_16x16x4_f32(false, a, false, b,
                                              (short)0, c, false, false);

    if (full) {
        #pragma unroll
        for (int r = 0; r < 8; ++r)
            h1[(size_t)(r0 + r + 8 * hi) * 16 + m] = c[r];
    } else {
        #pragma unroll
        for (int r = 0; r < 8; ++r) {
            int orow = r0 + r + 8 * hi;
            if (orow < n) h1[(size_t)orow * 16 + m] = c[r];
        }
    }
}

// agg1[i][j] = b1[j] + h1[i][j]*dinv[i]^2   (bias + self-loop message)
__global__ void k_init_agg1(const float* __restrict__ h1,
                            const float* __restrict__ dinv,
                            const float* __restrict__ b1,
                            float* __restrict__ agg1, int n) {
    int t = blockIdx.x * blockDim.x + threadIdx.x;
    if (t < n * 16) {
        int i = t >> 4, j = t & 15;
        float di = dinv[i];
        agg1[t] = b1[j] + h1[t] * di * di;
    }
}

// Edge scatter: agg1[dst] += h1[src] * (dinv[src]*dinv[dst]), 16 feats/edge.
// Grid-stride with next-iteration prefetch of the streaming int64 edge lists.
__global__ void k_scatter1(const long long* __restrict__ src,
                           const long long* __restrict__ dst,
                           const float* __restrict__ dinv,
                           const float* __restrict__ h1,
                           float* __restrict__ agg1, int E) {
    int stride = gridDim.x * blockDim.x;
    for (int e = blockIdx.x * blockDim.x + threadIdx.x; e < E; e += stride) {
        __builtin_prefetch(&src[e + stride], 0, 1);
        __builtin_prefetch(&dst[e + stride], 0, 1);
        int s = (int)src[e];
        int d = (int)dst[e];
        float nrm = dinv[s] * dinv[d];
        const float4* hp = (const float4*)(h1 + (size_t)s * 16);
        float* op = agg1 + (size_t)d * 16;
        #pragma unroll
        for (int q = 0; q < 4; ++q) {
            float4 v = hp[q];           // 128-bit L2-resident gather
            atomicAdd(op + q * 4 + 0, v.x * nrm);
            atomicAdd(op + q * 4 + 1, v.y * nrm);
            atomicAdd(op + q * 4 + 2, v.z * nrm);
            atomicAdd(op + q * 4 + 3, v.w * nrm);
        }
    }
}

// ---------------------------------------------------------------------------
// Layer 2 dense transform: h2[N,2] = relu(agg1)[N,16] @ W2[16,2]
// K=16 -> chain of four K=4 WMMAs accumulating through C. ReLU fused into A.
// ---------------------------------------------------------------------------
__global__ void k_gemm2_wmma(const float* __restrict__ agg1,
                             const float* __restrict__ W2,
                             float* __restrict__ h2, int n) {
    int gtid = blockIdx.x * blockDim.x + threadIdx.x;
    int wave = gtid >> 5;
    int lane = threadIdx.x & 31;
    int numTiles = (n + 15) >> 4;
    if (wave >= numTiles) return;       // wave-uniform exit

    int r0 = wave << 4;
    int m  = lane & 15;
    int hi = lane >> 4;
    int row = r0 + m;
    bool full = (r0 + 16 <= n);
    int rr = row < n ? row : (n - 1);
    bool rok = (row < n);

    v8f c = {};
    #pragma unroll
    for (int k0 = 0; k0 < 16; k0 += 4) {
        int k = k0 + hi * 2;
        float2 av = *(const float2*)(agg1 + (size_t)rr * 16 + k);
        float wa = W2[(k + 0) * 2 + (m & 1)];
        float wb = W2[(k + 1) * 2 + (m & 1)];
        v2f a, b;
        a.x = rok ? fmaxf(av.x, 0.0f) : 0.0f;   // fused ReLU
        a.y = rok ? fmaxf(av.y, 0.0f) : 0.0f;
        b.x = (m < 2) ? wa : 0.0f;
        b.y = (m < 2) ? wb : 0.0f;
        c = __builtin_amdgcn_wmma_f32_16x16x4_f32(false, a, false, b,
                                                  (short)0, c, false, false);
    }

    if (m < 2) {
        if (full) {
            #pragma unroll
            for (int r = 0; r < 8; ++r)
                h2[(size_t)(r0 + r + 8 * hi) * 2 + m] = c[r];
        } else {
            #pragma unroll
            for (int r = 0; r < 8; ++r) {
                int orow = r0 + r + 8 * hi;
                if (orow < n) h2[(size_t)orow * 2 + m] = c[r];
            }
        }
    }
}

__global__ void k_init_agg2(const float* __restrict__ h2,
                            const float* __restrict__ dinv,
                            const float* __restrict__ b2,
                            float* __restrict__ agg2, int n) {
    int t = blockIdx.x * blockDim.x + threadIdx.x;
    if (t < n * 2) {
        int i = t >> 1, j = t & 1;
        float di = dinv[i];
        agg2[t] = b2[j] + h2[t] * di * di;
    }
}

__global__ void k_scatter2(const long long* __restrict__ src,
                           const long long* __restrict__ dst,
                           const float* __restrict__ dinv,
                           const float* __restrict__ h2,
                           float* __restrict__ agg2, int E) {
    int stride = gridDim.x * blockDim.x;
    for (int e = blockIdx.x * blockDim.x + threadIdx.x; e < E; e += stride) {
        __builtin_prefetch(&src[e + stride], 0, 1);
        __builtin_prefetch(&dst[e + stride], 0, 1);
        int s = (int)src[e];
        int d = (int)dst[e];
        float nrm = dinv[s] * dinv[d];
        float2 v = *(const float2*)(h2 + (size_t)s * 2);
        atomicAdd(agg2 + (size_t)d * 2 + 0, v.x * nrm);
        atomicAdd(agg2 + (size_t)d * 2 + 1, v.y * nrm);
    }
}

// Row-wise log_softmax over 2 classes
__global__ void k_logsoftmax2(const float* __restrict__ agg2,
                              float* __restrict__ out, int n) {
    int i = blockIdx.x * blockDim.x + threadIdx.x;
    if (i >= n) return;
    float a = agg2[i * 2 + 0];
    float b = agg2[i * 2 + 1];
    float mx = fmaxf(a, b);
    float lse = mx + logf(expf(a - mx) + expf(b - mx));
    out[i * 2 + 0] = a - lse;
    out[i * 2 + 1] = b - lse;
}

// ---------------------------------------------------------------------------
// Launch
// ---------------------------------------------------------------------------
extern "C" void kernel_launch(void* const* d_in, const int* in_sizes, int n_in,
                              void* d_out, int out_size, void* d_ws, size_t ws_size,
                              hipStream_t stream) {
    const float*     x  = (const float*)d_in[0];
    const long long* ei = (const long long*)d_in[1];   // int64 per reference
    const float*     W1 = (const float*)d_in[2];
    const float*     b1 = (const float*)d_in[3];
    const float*     W2 = (const float*)d_in[4];
    const float*     b2 = (const float*)d_in[5];
    float*           out = (float*)d_out;

    const int N = in_sizes[0] / 2;
    const int E = in_sizes[1] / 2;
    const long long* src = ei;
    const long long* dst = ei + E;

    // Workspace layout (floats), regions padded to 16B alignment.
    float* ws   = (float*)d_ws;
    size_t off  = 0;
    auto alloc  = [&](size_t nf) { float* p = ws + off; off += (nf + 3) & ~size_t(3); return p; };
    float* deg  = alloc((size_t)N);
    float* dinv = alloc((size_t)N);
    float* h1   = alloc((size_t)N * 16);
    float* agg1 = alloc((size_t)N * 16);
    float* h2   = alloc((size_t)N * 2);
    float* agg2 = alloc((size_t)N * 2);

    const int B = 256;
    int gN   = (N + B - 1) / B;
    int gN16 = (N * 16 + B - 1) / B;
    int gN2  = (N * 2 + B - 1) / B;
    int tiles = (N + 15) / 16;
    int gT   = (tiles + 7) / 8;        // 8 waves per 256-thread block

    // Oversubscribed fixed grid for grid-stride edge loops (~6 iters/thread
    // at E=3.2M) so the next-iteration prefetch has a stable distance.
    int gEdge = 2048;
    if ((long long)gEdge * B > E) gEdge = (E + B - 1) / B;  // small-E fallback

    // degree + normalization
    k_init_deg <<<gN, B, 0, stream>>>(deg, N);
    k_deg_edges<<<gEdge, B, 0, stream>>>(dst, deg, E);
    k_dinv     <<<gN, B, 0, stream>>>(deg, dinv, N);

    // layer 1
    k_gemm1_wmma<<<gT, B, 0, stream>>>(x, W1, h1, N);
    k_init_agg1 <<<gN16, B, 0, stream>>>(h1, dinv, b1, agg1, N);
    k_scatter1  <<<gEdge, B, 0, stream>>>(src, dst, dinv, h1, agg1, E);

    // layer 2 (ReLU fused into GEMM A-operand)
    k_gemm2_wmma<<<gT, B, 0, stream>>>(agg1, W2, h2, N);
    k_init_agg2 <<<gN2, B, 0, stream>>>(h2, dinv, b2, agg2, N);
    k_scatter2  <<<gEdge, B, 0, stream>>>(src, dst, dinv, h2, agg2, E);

    // log-softmax
    k_logsoftmax2<<<gN, B, 0, stream>>>(agg2, out, N);
}